// RoIPooling_84387517431933
// MI455X (gfx1250) — compile-verified
//
#include <hip/hip_runtime.h>
#include <hip/hip_bf16.h>
#include <stdint.h>

// RoI max pooling for MI455X (gfx1250).
//
// Memory-bound (~44 MB read / 12.8 MB write -> ~2.5us at 23.3 TB/s; the
// 16.8 MB feature map is L2-resident in the 192 MB L2). WMMA cannot express
// a max-reduction, so the CDNA5-specific path is the async global->LDS copy
// engine (global_load_async_to_lds_b32, ASYNCcnt, s_wait_asynccnt): each
// block stages its RoI window exactly once into LDS with wave-coalesced
// per-lane copies, then the overlapping 7x7 bins re-read LDS instead of
// re-reading global memory.
//
// Dims are fixed by the reference (N=4, C=256, H=W=64, pooled 7x7) and
// hardcoded so all index math lowers to shifts/masks instead of the
// magic-number division chains seen in the previous round's assembly.

#define C_DIM       256
#define H_DIM       64
#define W_DIM       64
#define POOLED      7
#define BINS        49    // 7*7
#define CH_PER_BLK  8     // channels staged per block
#define TILE_H      32    // max window rows held in LDS
#define TILE_W      33    // padded row stride (32 cols + 1 pad vs bank conflicts)
#define THREADS     256   // 8 wave32s
#define NWAVES      8

__device__ __forceinline__ int clampi(int v, int lo, int hi) {
    return v < lo ? lo : (v > hi ? hi : v);
}

__global__ __launch_bounds__(THREADS)
void RoIPooling_84387517431933_kernel(const float* __restrict__ feats,
                                      const float* __restrict__ rois,
                                      float* __restrict__ out) {
    __shared__ float tile[CH_PER_BLK * TILE_H * TILE_W];   // 33 KB

    const int k    = blockIdx.x >> 5;                      // 32 chunks per roi
    const int c0   = (blockIdx.x & 31) << 3;               // 8 channels per chunk
    const int tid  = threadIdx.x;
    const int wid  = tid >> 5;                             // wave id   0..7
    const int lane = tid & 31;                             // lane id   0..31

    // ---- RoI geometry (registers, identical across the block) ----
    const float* roi = rois + (size_t)k * 5;
    const int b  = (int)roi[0];
    const float sc = 0.0625f;                              // 1/16
    const int x1 = (int)rintf(roi[1] * sc);                // rintf == jnp.round (half-even)
    const int y1 = (int)rintf(roi[2] * sc);
    const int x2 = (int)rintf(roi[3] * sc);
    const int y2 = (int)rintf(roi[4] * sc);
    int rh = y2 - y1 + 1; if (rh < 1) rh = 1;
    int rw = x2 - x1 + 1; if (rw < 1) rw = 1;
    const float bin_h = (float)rh / 7.0f;
    const float bin_w = (float)rw / 7.0f;

    // Window bounding box = union of all 7x7 bins (reference clip semantics)
    const int h0 = clampi(y1, 0, H_DIM);                                  // hstart[0]
    const int h1 = clampi((int)ceilf(7.0f * bin_h) + y1, 0, H_DIM);       // hend[6]
    const int w0 = clampi(x1, 0, W_DIM);                                  // wstart[0]
    const int w1 = clampi((int)ceilf(7.0f * bin_w) + x1, 0, W_DIM);       // wend[6]
    const int WH = h1 - h0;
    const int WW = w1 - w0;

    const bool use_lds = (WH > 0) && (WW > 0) && (WH <= TILE_H) && (WW <= 32);

    // ---- Stage window for 8 channels into LDS via async copy ----
    // c unrolled, rows strided by wave, cols strided by lane: zero runtime
    // divisions, and lane->w contiguity makes each wave's async gather a
    // coalesced row segment.
    if (use_lds) {
        const float* gbase = feats +
            (((size_t)b * C_DIM + c0) * H_DIM + h0) * W_DIM + w0;
#pragma unroll
        for (int c = 0; c < CH_PER_BLK; ++c) {
            const float* gch = gbase + (size_t)c * (H_DIM * W_DIM);
            float* tch = &tile[c * TILE_H * TILE_W];
            for (int h = wid; h < WH; h += NWAVES) {
                const float* grow = gch + (size_t)h * W_DIM;
                float* trow = tch + h * TILE_W;
                for (int w = lane; w < WW; w += 32) {
                    const uint32_t lds_addr =
                        (uint32_t)(size_t)(const void*)&trow[w];
                    // copies one dword from global memory straight into LDS; tracked by ASYNCcnt, no VGPR round trip.
                    asm volatile("global_load_async_to_lds_b32 %0, %1, off"
                                 :: "v"(lds_addr), "v"(grow + w)
                                 : "memory");
                }
            }
        }
#if __has_builtin(__builtin_amdgcn_s_wait_asynccnt)
        __builtin_amdgcn_s_wait_asynccnt(0);
#else
        asm volatile("s_wait_asynccnt 0x0" ::: "memory");
#endif
        __syncthreads();   // every wave drained its own ASYNCcnt -> LDS coherent
    }

    // ---- Compute: 8 ch * 49 bins = 392 outputs, 2 strided slots per thread ----
#pragma unroll
    for (int slot = 0; slot < 2; ++slot) {
        const int o = tid + slot * THREADS;
        if (o >= CH_PER_BLK * BINS) break;
        const int c   = o / BINS;          // constant divisor -> mulhi, no rcp chain
        const int bin = o - c * BINS;
        const int ph  = bin / POOLED;
        const int pw  = bin - ph * POOLED;

        const int hs = clampi((int)floorf((float)ph * bin_h) + y1, 0, H_DIM);
        const int he = clampi((int)ceilf(((float)ph + 1.0f) * bin_h) + y1, 0, H_DIM);
        const int ws = clampi((int)floorf((float)pw * bin_w) + x1, 0, W_DIM);
        const int we = clampi((int)ceilf(((float)pw + 1.0f) * bin_w) + x1, 0, W_DIM);

        float acc = -1e30f;   // NEG sentinel, matches reference
        if (use_lds) {
            const float* tc = &tile[c * TILE_H * TILE_W];
            for (int h = hs; h < he; ++h) {
                const float* trow = tc + (h - h0) * TILE_W - w0;
                for (int w = ws; w < we; ++w)
                    acc = fmaxf(acc, trow[w]);
            }
        } else {
            // Fallback for windows larger than the LDS tile; also covers
            // fully-clipped RoIs (empty loops -> sentinel -> 0 below).
            const float* fch = feats + ((size_t)b * C_DIM + (c0 + c)) * (H_DIM * W_DIM);
            for (int h = hs; h < he; ++h)
                for (int w = ws; w < we; ++w)
                    acc = fmaxf(acc, fch[h * W_DIM + w]);
        }
        // Empty bin => sentinel => 0 (Caffe semantics, same as reference mask)
        out[(((size_t)k * C_DIM + (c0 + c)) * BINS) + bin] =
            (acc <= -0.5e30f) ? 0.0f : acc;
    }
}

extern "C" void kernel_launch(void* const* d_in, const int* in_sizes, int n_in,
                              void* d_out, int out_size, void* d_ws, size_t ws_size,
                              hipStream_t stream) {
    const float* feats = (const float*)d_in[0];   // [4, 256, 64, 64] f32
    const float* rois  = (const float*)d_in[1];   // [K, 5] f32
    float* out = (float*)d_out;                   // [K, 256, 7, 7] f32

    const int K = in_sizes[1] / 5;

    dim3 grid(K * (C_DIM / CH_PER_BLK));          // 256 rois * 32 chunks = 8192 blocks
    RoIPooling_84387517431933_kernel<<<grid, THREADS, 0, stream>>>(feats, rois, out);
}